// LinearAttention_15281493639646
// MI455X (gfx1250) — compile-verified
//
#include <hip/hip_runtime.h>

typedef __attribute__((ext_vector_type(2))) float v2f;
typedef __attribute__((ext_vector_type(8))) float v8f;

#define NPIX 4096
#define QKV_ROWS 1536
#define NSPLIT 8   // K-dim split for the context stage

__device__ __forceinline__ v8f wmma_f32(v2f a, v2f b, v8f c) {
    // D = A(16x4) * B(4x16) + C(16x16), full fp32
    return __builtin_amdgcn_wmma_f32_16x16x4_f32(
        /*neg_a=*/false, a, /*neg_b=*/false, b,
        /*c_mod=*/(short)0, c, /*reuse_a=*/false, /*reuse_b=*/false);
}

// ---------------------------------------------------------------------------
// Generic NN GEMM: C[z] (MxN) = A (MxK, shared) @ B[z] (KxN) [+ bias[row]]
// Block tile 128x128, KT=32, 256 threads = 8 waves arranged 4(M) x 2(N),
// wave tile 32x64 = 2x4 WMMA accumulators.
// Both LDS tiles are k-contiguous per row so every fragment is one b64 load.
// Next K-chunk is register-prefetched under the WMMA inner loop.
// ---------------------------------------------------------------------------
__global__ __launch_bounds__(256)
void gemm_nn_kernel(const float* __restrict__ A, const float* __restrict__ B,
                    float* __restrict__ C, const float* __restrict__ bias,
                    int M, int N, int K, long long sB, long long sC)
{
    __shared__ float As[128][34];   // [m][k], 136B row stride: 8B aligned
    __shared__ float Bs[128][34];   // [n][k] (transposed), same stride

    const int tid  = threadIdx.x;
    const int lane = tid & 31, lr = lane & 15, lh = lane >> 4;
    const int wid  = tid >> 5;
    const int wm   = wid >> 1, wn = wid & 1;
    const int bm   = blockIdx.y * 128, bn = blockIdx.x * 128;
    const float* Bb = B + (long long)blockIdx.z * sB;
    float*       Cb = C + (long long)blockIdx.z * sC;

    v8f zero = {};
    v8f acc[2][4];
    for (int i = 0; i < 2; i++)
        for (int j = 0; j < 4; j++) acc[i][j] = zero;

    float4 pa[4], pb[4];
#pragma unroll
    for (int i = 0; i < 4; i++) {
        int idx = (tid + i * 256) * 4;
        int ra = idx >> 5, ca = idx & 31;
        pa[i] = *(const float4*)(A + (long long)(bm + ra) * K + ca);
        int rb = idx >> 7, cb = idx & 127;
        pb[i] = *(const float4*)(Bb + (long long)rb * N + bn + cb);
    }

    for (int kb = 0; kb < K; kb += 32) {
        // commit prefetched tiles to LDS (B transposed)
#pragma unroll
        for (int i = 0; i < 4; i++) {
            int idx = (tid + i * 256) * 4;
            int ra = idx >> 5, ca = idx & 31;
            As[ra][ca]   = pa[i].x; As[ra][ca+1] = pa[i].y;
            As[ra][ca+2] = pa[i].z; As[ra][ca+3] = pa[i].w;
            int rb = idx >> 7, cb = idx & 127;
            Bs[cb  ][rb] = pb[i].x; Bs[cb+1][rb] = pb[i].y;
            Bs[cb+2][rb] = pb[i].z; Bs[cb+3][rb] = pb[i].w;
        }
        __syncthreads();

        // prefetch next K-chunk while WMMAs below execute
        int kn = kb + 32;
        if (kn < K) {
#pragma unroll
            for (int i = 0; i < 4; i++) {
                int idx = (tid + i * 256) * 4;
                int ra = idx >> 5, ca = idx & 31;
                pa[i] = *(const float4*)(A + (long long)(bm + ra) * K + kn + ca);
                int rb = idx >> 7, cb = idx & 127;
                pb[i] = *(const float4*)(Bb + (long long)(kn + rb) * N + bn + cb);
            }
        }

#pragma unroll
        for (int k0 = 0; k0 < 32; k0 += 4) {
            v2f af[2], bf[4];
#pragma unroll
            for (int mi = 0; mi < 2; mi++)
                af[mi] = *(const v2f*)&As[wm * 32 + mi * 16 + lr][k0 + lh * 2];
#pragma unroll
            for (int ni = 0; ni < 4; ni++)
                bf[ni] = *(const v2f*)&Bs[wn * 64 + ni * 16 + lr][k0 + lh * 2];
#pragma unroll
            for (int mi = 0; mi < 2; mi++)
#pragma unroll
                for (int ni = 0; ni < 4; ni++)
                    acc[mi][ni] = wmma_f32(af[mi], bf[ni], acc[mi][ni]);
        }
        __syncthreads();
    }

#pragma unroll
    for (int mi = 0; mi < 2; mi++)
#pragma unroll
        for (int ni = 0; ni < 4; ni++) {
            int col = bn + wn * 64 + ni * 16 + lr;
#pragma unroll
            for (int r = 0; r < 8; r++) {
                int row = bm + wm * 32 + mi * 16 + lh * 8 + r;
                float v = acc[mi][ni][r];
                if (bias) v += bias[row];
                Cb[(long long)row * N + col] = v;
            }
        }
}

// ---------------------------------------------------------------------------
// Row-wise stable softmax over n=4096, in place on the K section of qkv.
// ---------------------------------------------------------------------------
__global__ __launch_bounds__(256)
void softmax_rows_kernel(float* __restrict__ qkv)
{
    const int row = blockIdx.x;          // 0..4095
    const int b = row >> 9, r = row & 511;
    float* p = qkv + ((long long)b * QKV_ROWS + 512 + r) * NPIX;

    __shared__ float red[256];
    const int tid = threadIdx.x;

    float m = -3.4e38f;
    for (int i = tid; i < NPIX; i += 256) m = fmaxf(m, p[i]);
    red[tid] = m; __syncthreads();
    for (int s = 128; s > 0; s >>= 1) {
        if (tid < s) red[tid] = fmaxf(red[tid], red[tid + s]);
        __syncthreads();
    }
    m = red[0]; __syncthreads();

    float sum = 0.f;
    for (int i = tid; i < NPIX; i += 256) {
        float e = __expf(p[i] - m);
        p[i] = e; sum += e;
    }
    red[tid] = sum; __syncthreads();
    for (int s = 128; s > 0; s >>= 1) {
        if (tid < s) red[tid] += red[tid + s];
        __syncthreads();
    }
    float inv = 1.f / red[0];
    for (int i = tid; i < NPIX; i += 256) p[i] *= inv;
}

// ---------------------------------------------------------------------------
// Partial contextT[e][d] = sum_{n in split} V[e,n] * K[d,n]
// grid (bh=64, split=NSPLIT); each split covers 4096/NSPLIT n-values.
// 8 waves 2(M)x4(N), wave tile 32x16. Both fragments contiguous b64 reads.
// ---------------------------------------------------------------------------
__global__ __launch_bounds__(256)
void context_kernel(const float* __restrict__ qkv, float* __restrict__ ctxp)
{
    const int bh = blockIdx.x;
    const int b = bh >> 3, h = bh & 7;
    const float* Kp = qkv + ((long long)b * QKV_ROWS +  512 + h * 64) * NPIX;
    const float* Vp = qkv + ((long long)b * QKV_ROWS + 1024 + h * 64) * NPIX;

    __shared__ float Vs[64][66];
    __shared__ float Ks[64][66];

    const int tid  = threadIdx.x;
    const int lane = tid & 31, lr = lane & 15, lh = lane >> 4;
    const int wid  = tid >> 5;
    const int wm   = wid >> 2, wn = wid & 3;

    v8f zero = {};
    v8f acc[2]; acc[0] = zero; acc[1] = zero;

    const int n0 = blockIdx.y * (NPIX / NSPLIT);
    const int n1 = n0 + (NPIX / NSPLIT);

    float4 pv[4], pk[4];
#pragma unroll
    for (int i = 0; i < 4; i++) {
        int idx = (tid + i * 256) * 4;
        int r = idx >> 6, c = idx & 63;
        pv[i] = *(const float4*)(Vp + (long long)r * NPIX + n0 + c);
        pk[i] = *(const float4*)(Kp + (long long)r * NPIX + n0 + c);
    }

    for (int nb = n0; nb < n1; nb += 64) {
#pragma unroll
        for (int i = 0; i < 4; i++) {
            int idx = (tid + i * 256) * 4;
            int r = idx >> 6, c = idx & 63;
            Vs[r][c] = pv[i].x; Vs[r][c+1] = pv[i].y; Vs[r][c+2] = pv[i].z; Vs[r][c+3] = pv[i].w;
            Ks[r][c] = pk[i].x; Ks[r][c+1] = pk[i].y; Ks[r][c+2] = pk[i].z; Ks[r][c+3] = pk[i].w;
        }
        __syncthreads();

        int nn = nb + 64;
        if (nn < n1) {
#pragma unroll
            for (int i = 0; i < 4; i++) {
                int idx = (tid + i * 256) * 4;
                int r = idx >> 6, c = idx & 63;
                pv[i] = *(const float4*)(Vp + (long long)r * NPIX + nn + c);
                pk[i] = *(const float4*)(Kp + (long long)r * NPIX + nn + c);
            }
        }

#pragma unroll
        for (int k0 = 0; k0 < 64; k0 += 4) {
            v2f bfr = *(const v2f*)&Ks[wn * 16 + lr][k0 + lh * 2];
#pragma unroll
            for (int mi = 0; mi < 2; mi++) {
                v2f afr = *(const v2f*)&Vs[wm * 32 + mi * 16 + lr][k0 + lh * 2];
                acc[mi] = wmma_f32(afr, bfr, acc[mi]);
            }
        }
        __syncthreads();
    }

    float* Cp = ctxp + ((long long)blockIdx.y * 64 + bh) * 4096;  // CT[e][d]
#pragma unroll
    for (int mi = 0; mi < 2; mi++) {
        int d = wn * 16 + lr;
#pragma unroll
        for (int r = 0; r < 8; r++) {
            int e = wm * 32 + mi * 16 + lh * 8 + r;
            Cp[e * 64 + d] = acc[mi][r];
        }
    }
}

// Sum the NSPLIT partial contexts: ctx[i] = sum_p ctxp[p][i]
__global__ __launch_bounds__(256)
void ctx_reduce_kernel(const float* __restrict__ ctxp, float* __restrict__ ctx)
{
    const long long i = (long long)blockIdx.x * 256 + threadIdx.x;  // 0..262143
    float s = 0.f;
#pragma unroll
    for (int p = 0; p < NSPLIT; p++)
        s += ctxp[(long long)p * 64 * 4096 + i];
    ctx[i] = s;
}

// ---------------------------------------------------------------------------
// attn_out[h*64+e][n] = sum_d CT[e][d] * Q[d][n]  (NN, M=64, K=64, N=4096)
// grid (n-chunks of 128, b*8+h). 8 waves 2(M)x4(N), wave tile 32x32.
// Q tile stored transposed in LDS -> contiguous b64 fragment loads.
// ---------------------------------------------------------------------------
__global__ __launch_bounds__(256)
void attn_out_kernel(const float* __restrict__ qkv, const float* __restrict__ ctx,
                     float* __restrict__ attn)
{
    const int bh = blockIdx.y;
    const int b = bh >> 3, h = bh & 7;
    const int bn = blockIdx.x * 128;
    const float* Qp = qkv  + ((long long)b * QKV_ROWS + h * 64) * NPIX;
    const float* Cp = ctx  + (long long)bh * 4096;
    float*       Op = attn + ((long long)b * 512 + h * 64) * NPIX;

    __shared__ float CTs[64][66];    // [e][d]
    __shared__ float Qs[128][66];    // [n][d] (transposed)

    const int tid  = threadIdx.x;
    const int lane = tid & 31, lr = lane & 15, lh = lane >> 4;
    const int wid  = tid >> 5;
    const int wm   = wid >> 2, wn = wid & 3;

#pragma unroll
    for (int i = 0; i < 4; i++) {
        int idx = (tid + i * 256) * 4;
        int r = idx >> 6, c = idx & 63;
        float4 v = *(const float4*)(Cp + r * 64 + c);
        CTs[r][c] = v.x; CTs[r][c+1] = v.y; CTs[r][c+2] = v.z; CTs[r][c+3] = v.w;
    }
#pragma unroll
    for (int i = 0; i < 8; i++) {
        int idx = (tid + i * 256) * 4;
        int r = idx >> 7, c = idx & 127;            // r = d-row, c = n-col
        float4 v = *(const float4*)(Qp + (long long)r * NPIX + bn + c);
        Qs[c  ][r] = v.x; Qs[c+1][r] = v.y; Qs[c+2][r] = v.z; Qs[c+3][r] = v.w;
    }
    __syncthreads();

    v8f zero = {};
    v8f acc[2][2];
    acc[0][0] = zero; acc[0][1] = zero; acc[1][0] = zero; acc[1][1] = zero;

#pragma unroll
    for (int k0 = 0; k0 < 64; k0 += 4) {
        v2f af[2], bf[2];
#pragma unroll
        for (int mi = 0; mi < 2; mi++)
            af[mi] = *(const v2f*)&CTs[wm * 32 + mi * 16 + lr][k0 + lh * 2];
#pragma unroll
        for (int ni = 0; ni < 2; ni++)
            bf[ni] = *(const v2f*)&Qs[wn * 32 + ni * 16 + lr][k0 + lh * 2];
#pragma unroll
        for (int mi = 0; mi < 2; mi++)
#pragma unroll
            for (int ni = 0; ni < 2; ni++)
                acc[mi][ni] = wmma_f32(af[mi], bf[ni], acc[mi][ni]);
    }

#pragma unroll
    for (int mi = 0; mi < 2; mi++)
#pragma unroll
        for (int ni = 0; ni < 2; ni++) {
            int n = bn + wn * 32 + ni * 16 + lr;
#pragma unroll
            for (int r = 0; r < 8; r++) {
                int e = wm * 32 + mi * 16 + lh * 8 + r;
                Op[(long long)e * NPIX + n] = acc[mi][ni][r];
            }
        }
}

// ---------------------------------------------------------------------------
extern "C" void kernel_launch(void* const* d_in, const int* in_sizes, int n_in,
                              void* d_out, int out_size, void* d_ws, size_t ws_size,
                              hipStream_t stream)
{
    const float* x     = (const float*)d_in[0];   // (8, 512, 64, 64)
    const float* w_qkv = (const float*)d_in[1];   // (1536, 512)
    const float* w_out = (const float*)d_in[2];   // (512, 512)
    const float* b_out = (const float*)d_in[3];   // (512,)
    float* out = (float*)d_out;                   // (8, 512, 64, 64)

    float* qkv  = (float*)d_ws;                              // 8*1536*4096 f32
    float* attn = qkv  + (long long)8 * QKV_ROWS * NPIX;     // 8*512*4096 f32
    float* ctx  = attn + (long long)8 * 512 * NPIX;          // 8*8*64*64 f32
    float* ctxp = ctx  + (long long)64 * 4096;               // NSPLIT partials

    // 1) qkv[b,o,p] = sum_c w_qkv[o,c] * x[b,c,p]
    gemm_nn_kernel<<<dim3(32, 12, 8), 256, 0, stream>>>(
        w_qkv, x, qkv, nullptr, QKV_ROWS, NPIX, 512,
        (long long)512 * NPIX, (long long)QKV_ROWS * NPIX);

    // 2) softmax over n on K rows (in place)
    softmax_rows_kernel<<<4096, 256, 0, stream>>>(qkv);

    // 3) contextT[e,d] per (b,h), split over n then reduced
    context_kernel<<<dim3(64, NSPLIT), 256, 0, stream>>>(qkv, ctxp);
    ctx_reduce_kernel<<<(64 * 4096) / 256, 256, 0, stream>>>(ctxp, ctx);

    // 4) attn_out = contextT @ Q per (b,h)
    attn_out_kernel<<<dim3(32, 64), 256, 0, stream>>>(qkv, ctx, attn);

    // 5) out[b,o,p] = sum_c w_out[o,c] * attn[b,c,p] + b_out[o]
    gemm_nn_kernel<<<dim3(32, 4, 8), 256, 0, stream>>>(
        w_out, attn, out, b_out, 512, NPIX, 512,
        (long long)512 * NPIX, (long long)512 * NPIX);
}